// RadiologicalDepthLayer_19524921327957
// MI455X (gfx1250) — compile-verified
//
#include <hip/hip_runtime.h>
#include <hip/hip_bf16.h>

// Problem constants (from reference setup_inputs)
#define BB 4
#define HH 128
#define DD 512
#define WW 512
#define GG 72
#define PP 65536

typedef float v2f __attribute__((ext_vector_type(2)));
typedef float v8f __attribute__((ext_vector_type(8)));

// Async-to-LDS builtin (ROCm 7.2 clang-22) takes pointers to gcc-vector int4:
//   arg0: AS1 (global) v4i*,  arg1: AS3 (LDS) v4i*,  arg2/arg3: imm offset/cpol
typedef int v4i __attribute__((vector_size(16)));
typedef __attribute__((address_space(1))) v4i* gptr_v4i;
typedef __attribute__((address_space(3))) v4i* lptr_v4i;

#if __has_builtin(__builtin_amdgcn_global_load_async_to_lds_b128)
#define USE_ASYNC_LDS 1
#else
#define USE_ASYNC_LDS 0
#endif

#if USE_ASYNC_LDS
#if __has_builtin(__builtin_amdgcn_s_wait_asynccnt)
#define WAIT_ASYNC(n) __builtin_amdgcn_s_wait_asynccnt(n)
#else
#define WAIT_ASYNC(n) asm volatile("s_wait_asynccnt %0" ::"n"(n))
#endif
#else
#define WAIT_ASYNC(n)
#endif

__device__ __forceinline__ float readlane_f(float v, int l) {
    return __uint_as_float(__builtin_amdgcn_readlane(__float_as_uint(v), l));
}

// ---------------------------------------------------------------------------
// Kernel 0: per-angle mean physical step size.  steps[g] = mean over p of
//   || (c[g,p+1]-c[g,p]) * RES ||,  RES = (2,2,2)  ->  2*sqrt(dx^2+dy^2+dz^2)
// ---------------------------------------------------------------------------
__global__ void __launch_bounds__(256) step_kernel(const float* __restrict__ coords,
                                                   float* __restrict__ steps)
{
    const int g = blockIdx.x;
    const float* c = coords + (size_t)g * PP * 3;
    float acc = 0.0f;
    for (int p = threadIdx.x; p < PP - 1; p += 256) {
        const float dx = c[(p + 1) * 3 + 0] - c[p * 3 + 0];
        const float dy = c[(p + 1) * 3 + 1] - c[p * 3 + 1];
        const float dz = c[(p + 1) * 3 + 2] - c[p * 3 + 2];
        acc += sqrtf(dx * dx + dy * dy + dz * dz);
    }
    __shared__ float red[256];
    red[threadIdx.x] = acc;
    __syncthreads();
    for (int off = 128; off > 0; off >>= 1) {
        if (threadIdx.x < off) red[threadIdx.x] += red[threadIdx.x + off];
        __syncthreads();
    }
    if (threadIdx.x == 0) steps[g] = red[0] * (2.0f / (float)(PP - 1));
}

// ---------------------------------------------------------------------------
// Kernel 1: trilinear gather.  One thread per (g,p); weights/indices computed
// once and reused across the 4 batch volumes.  Densities land in d_out rows
// (b*G+g)*P + p; the scan kernel then rewrites d_out in place.
// ---------------------------------------------------------------------------
__global__ void __launch_bounds__(256) density_kernel(const float* __restrict__ ct,
                                                      const float* __restrict__ coords,
                                                      float* __restrict__ dens)
{
    const unsigned idx = blockIdx.x * 256u + threadIdx.x;   // g*P + p
    if (idx >= (unsigned)GG * PP) return;
    const unsigned g = idx >> 16;          // P == 65536
    const unsigned p = idx & (PP - 1);

    float cx = coords[(size_t)idx * 3 + 0];
    float cy = coords[(size_t)idx * 3 + 1];
    float cz = coords[(size_t)idx * 3 + 2];
    cx = fminf(fmaxf(cx, 0.0f), (float)(WW - 1));
    cy = fminf(fmaxf(cy, 0.0f), (float)(DD - 1));
    cz = fminf(fmaxf(cz, 0.0f), (float)(HH - 1));

    const int x0 = (int)floorf(cx);
    const int y0 = (int)floorf(cy);
    const int z0 = (int)floorf(cz);
    const int x1 = min(x0 + 1, WW - 1);
    const int y1 = min(y0 + 1, DD - 1);
    const int z1 = min(z0 + 1, HH - 1);
    const float xd = cx - (float)x0;
    const float yd = cy - (float)y0;
    const float zd = cz - (float)z0;

    const size_t DW = (size_t)DD * WW;
    const size_t HDW = (size_t)HH * DW;
    const size_t b00 = (size_t)z0 * DW + (size_t)y0 * WW;
    const size_t b01 = (size_t)z0 * DW + (size_t)y1 * WW;
    const size_t b10 = (size_t)z1 * DW + (size_t)y0 * WW;
    const size_t b11 = (size_t)z1 * DW + (size_t)y1 * WW;

    const float omx = 1.0f - xd, omy = 1.0f - yd, omz = 1.0f - zd;

    for (int b = 0; b < BB; ++b) {
        const float* __restrict__ v = ct + (size_t)b * HDW;
        if (b + 1 < BB) {  // hide latency of the next volume's lines
            const float* nv = ct + (size_t)(b + 1) * HDW;
            __builtin_prefetch(nv + b00 + x0, 0, 1);
            __builtin_prefetch(nv + b01 + x0, 0, 1);
            __builtin_prefetch(nv + b10 + x0, 0, 1);
            __builtin_prefetch(nv + b11 + x0, 0, 1);
        }
        const float c000 = v[b00 + x0], c001 = v[b00 + x1];
        const float c010 = v[b01 + x0], c011 = v[b01 + x1];
        const float c100 = v[b10 + x0], c101 = v[b10 + x1];
        const float c110 = v[b11 + x0], c111 = v[b11 + x1];

        const float c00 = c000 * omx + c001 * xd;
        const float c01 = c010 * omx + c011 * xd;
        const float c10 = c100 * omx + c101 * xd;
        const float c11 = c110 * omx + c111 * xd;
        const float c0 = c00 * omy + c01 * yd;
        const float c1 = c10 * omy + c11 * yd;
        const float d = c0 * omz + c1 * zd;

        dens[((size_t)(b * GG + g)) * PP + p] = d;
    }
}

// ---------------------------------------------------------------------------
// Kernel 2: per-row inclusive scan, scaled:  out[p] = step*(incl[p] + 0.5*d[p])
// One workgroup (8 wave32) per row of 65536.  Double-buffered 16KB LDS chunks:
// chunk c+1 is staged with GLOBAL_LOAD_ASYNC_TO_LDS_B128 (ASYNCcnt) while
// chunk c is scanned.  Within a chunk: wave-segment totals -> exclusive wave
// offsets -> WMMA tile scans.  A 16x16 row-major tile T of 256 consecutive
// values is scanned as M1 = T x U (U = upper-triangular ones) via 4 chained
// V_WMMA_F32_16X16X4_F32; tile row sums (column 15, lanes 15/31) are pulled
// out with v_readlane and scalar-prefixed.
// ---------------------------------------------------------------------------
#define SCAN_CHUNK 4096                         // floats per chunk (16 KB)
#define N_CHUNKS   (PP / SCAN_CHUNK)            // 16
#define STAGE_OPS  (SCAN_CHUNK / 4 / 256)       // float4 loads per thread = 4

__global__ void __launch_bounds__(256) scan_kernel(float* __restrict__ data,
                                                   const float* __restrict__ steps)
{
    __shared__ float sbuf[2][SCAN_CHUNK];
    __shared__ float swave[16];   // [0..7] wave totals, [8..15] exclusive offsets

    const int tid  = threadIdx.x;
    const int lane = tid & 31;
    const int wave = tid >> 5;
    const int hi   = lane >> 4;     // which half of the wave (A/B K-pairing)
    const int m    = lane & 15;

    const int row = blockIdx.x;                 // row = b*G + g
    const float step = steps[row % GG];
    float* __restrict__ rp = data + (size_t)row * PP;

    // Constant B operand: slices of U (upper-triangular ones, incl. diagonal).
    // B layout: VGPR0 = rows {4s, 4s+2} (lanes 0-15 / 16-31), VGPR1 = rows {4s+1, 4s+3}.
    v2f bmat[4];
#pragma unroll
    for (int s = 0; s < 4; ++s) {
        bmat[s].x = ((4 * s + 2 * hi)     <= m) ? 1.0f : 0.0f;
        bmat[s].y = ((4 * s + 2 * hi + 1) <= m) ? 1.0f : 0.0f;
    }

    // Stage one chunk: async copy global -> LDS (fallback: plain copies).
    auto stage = [&](int chunk) {
        if (chunk >= N_CHUNKS) return;
        float4* dst4 = (float4*)sbuf[chunk & 1];
        const float4* src4 = (const float4*)(rp + chunk * SCAN_CHUNK);
#if USE_ASYNC_LDS
#pragma unroll
        for (int i = 0; i < STAGE_OPS; ++i) {
            __builtin_amdgcn_global_load_async_to_lds_b128(
                (gptr_v4i)(src4 + i * 256 + tid),
                (lptr_v4i)(dst4 + i * 256 + tid), 0, 0);
        }
#else
#pragma unroll
        for (int i = 0; i < STAGE_OPS; ++i)
            dst4[i * 256 + tid] = src4[i * 256 + tid];
#endif
    };

    stage(0);

    float rowCarry = 0.0f;
    for (int chunk = 0; chunk < N_CHUNKS; ++chunk) {
        stage(chunk + 1);   // overlap next chunk's staging with this chunk's scan
#if USE_ASYNC_LDS
        if (chunk + 1 < N_CHUNKS) { WAIT_ASYNC(STAGE_OPS); } else { WAIT_ASYNC(0); }
#endif
        __syncthreads();

        float* sdata = sbuf[chunk & 1];
        const int cbase = chunk * SCAN_CHUNK;

        // Phase A: per-wave segment totals (512 elements per wave).
        const int seg = wave * (SCAN_CHUNK / 8);
        float s = 0.0f;
#pragma unroll
        for (int i = 0; i < (SCAN_CHUNK / 8) / 32; ++i)
            s += sdata[seg + lane + 32 * i];
#pragma unroll
        for (int off = 16; off > 0; off >>= 1)
            s += __shfl_xor(s, off, 32);
        if (lane == 0) swave[wave] = s;
        __syncthreads();
        if (tid == 0) {
            float acc = 0.0f;
            for (int w = 0; w < 8; ++w) { swave[8 + w] = acc; acc += swave[w]; }
        }
        __syncthreads();

        float carry = rowCarry + swave[8 + wave];
        const float chunkTotal = swave[8 + 7] + swave[7];

        // Phase C: WMMA tile scans (2 tiles of 256 per wave).
#pragma unroll
        for (int t = 0; t < (SCAN_CHUNK / 8) / 256; ++t) {
            const int base = seg + t * 256;

            v8f acc = {};
#pragma unroll
            for (int sl = 0; sl < 4; ++sl) {
                // A layout (32-bit 16x4): VGPR0 = K {4sl, 4sl+2}, VGPR1 = K {4sl+1, 4sl+3}
                v2f a;
                const int col = 4 * sl + 2 * hi;
                a.x = sdata[base + 16 * m + col];
                a.y = sdata[base + 16 * m + col + 1];
                acc = __builtin_amdgcn_wmma_f32_16x16x4_f32(
                        false, a, false, bmat[sl], (short)0, acc, false, false);
            }

            // Tile row sums live in column 15 => lanes 15 (rows 0-7) / 31 (rows 8-15).
            float rs[16];
#pragma unroll
            for (int k = 0; k < 8; ++k) {
                rs[k]     = readlane_f(acc[k], 15);
                rs[k + 8] = readlane_f(acc[k], 31);
            }
            float pre[16];
            pre[0] = 0.0f;
#pragma unroll
            for (int r = 1; r < 16; ++r) pre[r] = pre[r - 1] + rs[r - 1];

#pragma unroll
            for (int k = 0; k < 8; ++k) {
                const float prer = hi ? pre[k + 8] : pre[k];   // row = k + 8*hi
                const int idx = base + 16 * k + 128 * hi + m;  // 16*row + col
                const float d = sdata[idx];
                const float y = acc[k] + prer + carry;
                rp[cbase + idx] = step * (y + 0.5f * d);
            }
            carry += pre[15] + rs[15];   // total of this tile
        }
        rowCarry += chunkTotal;
        __syncthreads();
    }
}

// ---------------------------------------------------------------------------
extern "C" void kernel_launch(void* const* d_in, const int* in_sizes, int n_in,
                              void* d_out, int out_size, void* d_ws, size_t ws_size,
                              hipStream_t stream) {
    const float* ct     = (const float*)d_in[0];   // [B,H,D,W] f32
    const float* coords = (const float*)d_in[1];   // [1,G,P,3] f32
    float* out          = (float*)d_out;           // [B*G, P] f32
    float* steps        = (float*)d_ws;            // 72 floats

    (void)in_sizes; (void)n_in; (void)out_size; (void)ws_size;

    step_kernel<<<GG, 256, 0, stream>>>(coords, steps);
    density_kernel<<<(GG * PP) / 256, 256, 0, stream>>>(ct, coords, out);
    scan_kernel<<<BB * GG, 256, 0, stream>>>(out, steps);
}